// VectorQuantizer_28114855919967
// MI455X (gfx1250) — compile-verified
//
#include <hip/hip_runtime.h>
#include <hip/hip_bf16.h>
#include <math.h>

typedef __attribute__((ext_vector_type(2))) float v2f;
typedef __attribute__((ext_vector_type(8))) float v8f;

#define K_EMB   1024
#define D_EMB   64
#define N_VEC   65536
#define ESTRIDE 68   // padded row stride (floats) -> conflict-free ds_load_b64

// ---------------------------------------------------------------------------
// ws layout (bytes):
//   [     0,   4096)  hist      : 1024 x u32
//   [  4096,  20480)  partial   : 4096 x f32   (per-block loss partial sums)
//   [ 20480,  24576)  enorm     : 1024 x f32   (||e_k||^2)
//   [ 24576, 286720)  idx       : 65536 x i32
// ---------------------------------------------------------------------------

// Kernel 1: embedding norms + zero histogram
__global__ void vq_prep(const float* __restrict__ emb,
                        float* __restrict__ enorm,
                        unsigned int* __restrict__ hist) {
    int k = blockIdx.x * blockDim.x + threadIdx.x;   // 1024 threads total
    if (k < K_EMB) {
        const float4* row = (const float4*)(emb + k * D_EMB);
        float s = 0.0f;
        #pragma unroll
        for (int j = 0; j < D_EMB / 4; ++j) {
            float4 v = row[j];
            s += v.x * v.x + v.y * v.y + v.z * v.z + v.w * v.w;
        }
        enorm[k] = s;
        hist[k]  = 0u;
    }
}

// Kernel 2: fused distance (WMMA fp32) + argmin + histogram
// grid: 512 blocks x 256 threads (8 waves). Each wave owns 16 rows of x.
__launch_bounds__(256)
__global__ void vq_argmin(const float* __restrict__ x,
                          const float* __restrict__ emb,
                          const float* __restrict__ enorm,
                          int* __restrict__ idx,
                          unsigned int* __restrict__ hist) {
    __shared__ float es[128 * ESTRIDE];   // 34816 B: one 128-row embedding chunk
    __shared__ float esn[128];            // chunk ||e||^2

    const int tid  = threadIdx.x;
    const int wave = tid >> 5;
    const int lane = tid & 31;
    const int lh   = lane & 15;   // lane within half
    const int hh   = lane >> 4;   // which half

    const int n0    = blockIdx.x * 128;       // 128 rows per WG, all in one batch b
    const int b     = n0 >> 10;               // H*W = 1024
    const int hwrow = (n0 & 1023) + wave * 16 + lh;
    const int xbase = b * 65536 + hwrow;      // NCHW: + c*1024 walks the channel

    // A fragments: a[d] = -2 * x for K-dims {4d+2hh, 4d+2hh+1}, row = lh.
    v2f a[16];
    #pragma unroll
    for (int d = 0; d < 16; ++d) {
        int c = 4 * d + 2 * hh;
        a[d].x = -2.0f * x[xbase + c * 1024];
        a[d].y = -2.0f * x[xbase + (c + 1) * 1024];
    }

    float best[8];
    int   bestk[8];
    #pragma unroll
    for (int v = 0; v < 8; ++v) { best[v] = 3.4e38f; bestk[v] = 0; }

    #pragma unroll 1
    for (int chunk = 0; chunk < 8; ++chunk) {
        const int krow0 = chunk * 128;
        __syncthreads();   // previous chunk fully consumed

        // --- async stage: 128 embedding rows, global -> LDS (no VGPR bounce).
        // IOFFSET applies to both the LDS destination and the global source.
        {
            const int r    = tid >> 1;     // 0..127
            const int half = tid & 1;      // 32-float half of the row
            const float*  gsrc = emb + (krow0 + r) * D_EMB + half * 32;
            unsigned int  ldsa = (unsigned int)(unsigned long long)
                                 (&es[r * ESTRIDE + half * 32]);
            asm volatile(
                "global_load_async_to_lds_b128 %0, %1, off\n\t"
                "global_load_async_to_lds_b128 %0, %1, off offset:16\n\t"
                "global_load_async_to_lds_b128 %0, %1, off offset:32\n\t"
                "global_load_async_to_lds_b128 %0, %1, off offset:48\n\t"
                "global_load_async_to_lds_b128 %0, %1, off offset:64\n\t"
                "global_load_async_to_lds_b128 %0, %1, off offset:80\n\t"
                "global_load_async_to_lds_b128 %0, %1, off offset:96\n\t"
                "global_load_async_to_lds_b128 %0, %1, off offset:112"
                :: "v"(ldsa), "v"(gsrc) : "memory");
            if (tid < 128) esn[tid] = enorm[krow0 + tid];   // norms via normal DS path
            asm volatile("s_wait_asynccnt 0x0" ::: "memory");
        }
        __syncthreads();

        // Pull next chunk toward the caches while this one computes.
        if (chunk < 7)
            __builtin_prefetch(emb + (krow0 + 128) * D_EMB + tid * 32, 0, 1);

        // --- compute: double-buffered B fragments AND enorm seed. The
        // sched_barrier(0) pins each tile's DS load burst (issued during the
        // previous tile's WMMAs) ahead of the WMMA chain, so every DS wait
        // covers loads issued a whole 16-WMMA tile earlier.
        v2f   bcur[16], bnext[16];
        float encur, ennext;
        {
            const float* bb0 = es + lh * ESTRIDE + 2 * hh;
            #pragma unroll
            for (int d = 0; d < 16; ++d) bcur[d] = *(const v2f*)(bb0 + 4 * d);
            encur = esn[lh];
        }
        __builtin_amdgcn_sched_barrier(0);
        #pragma unroll
        for (int kt = 0; kt < 8; ++kt) {
            if (kt < 7) {   // burst-issue next K-tile's B fragments + enorm seed
                const float* bbn = es + ((kt + 1) * 16 + lh) * ESTRIDE + 2 * hh;
                #pragma unroll
                for (int d = 0; d < 16; ++d) bnext[d] = *(const v2f*)(bbn + 4 * d);
                ennext = esn[(kt + 1) * 16 + lh];
            }
            __builtin_amdgcn_sched_barrier(0);   // loads stay above the WMMAs
            const int kglob = krow0 + kt * 16 + lh;
            v8f acc = {encur, encur, encur, encur, encur, encur, encur, encur};
            #pragma unroll
            for (int d = 0; d < 16; ++d) {
                acc = __builtin_amdgcn_wmma_f32_16x16x4_f32(
                        /*neg_a=*/false, a[d], /*neg_b=*/false, bcur[d],
                        /*c_mod=*/(short)0, acc, /*reuse_a=*/false, /*reuse_b=*/false);
            }
            // acc = ||e||^2 - 2 x.e  (row-constant ||x||^2 dropped: same argmin)
            #pragma unroll
            for (int v = 0; v < 8; ++v) {
                float dv = acc[v];
                if (dv < best[v]) { best[v] = dv; bestk[v] = kglob; }  // lowest k on ties
            }
            if (kt < 7) {
                #pragma unroll
                for (int d = 0; d < 16; ++d) bcur[d] = bnext[d];   // register rename
                encur = ennext;
            }
        }
    }

    // Min-reduce across the 16 lane-columns inside each half (ref tie-break: lowest k).
    #pragma unroll
    for (int off = 1; off < 16; off <<= 1) {
        #pragma unroll
        for (int v = 0; v < 8; ++v) {
            float od = __shfl_xor(best[v], off, 32);
            int   ok = __shfl_xor(bestk[v], off, 32);
            if (od < best[v] || (od == best[v] && ok < bestk[v])) {
                best[v] = od; bestk[v] = ok;
            }
        }
    }

    if (lh == 0) {
        #pragma unroll
        for (int v = 0; v < 8; ++v) {
            int n = n0 + wave * 16 + hh * 8 + v;   // C/D layout: VGPR v -> M = hh*8 + v
            idx[n] = bestk[v];
            atomicAdd(&hist[bestk[v]], 1u);
        }
    }
}

// Kernel 3: gather quantized output (NCHW, float4 coalesced) + loss partials
__launch_bounds__(256)
__global__ void vq_quant(const float* __restrict__ x,
                         const float* __restrict__ emb,
                         const int* __restrict__ idx,
                         float* __restrict__ outq,
                         float* __restrict__ partial) {
    __shared__ float bred[8];
    const int gid = blockIdx.x * 256 + threadIdx.x;   // 1,048,576 threads
    const int b   = gid >> 14;            // / (64 * 256)
    const int rem = gid & 16383;
    const int c   = rem >> 8;             // channel
    const int hw0 = (rem & 255) << 2;     // 4 consecutive spatial positions
    const int nb  = (b << 10) + hw0;
    const int off = b * 65536 + c * 1024 + hw0;  // same NCHW offset for in and out

    int4   i4 = *(const int4*)(idx + nb);
    float4 x4 = *(const float4*)(x + off);
    float4 q4;
    q4.x = emb[i4.x * 64 + c];
    q4.y = emb[i4.y * 64 + c];
    q4.z = emb[i4.z * 64 + c];
    q4.w = emb[i4.w * 64 + c];
    *(float4*)(outq + off) = q4;          // st estimator == q numerically

    float dx = q4.x - x4.x, dy = q4.y - x4.y, dz = q4.z - x4.z, dw = q4.w - x4.w;
    float s  = dx * dx + dy * dy + dz * dz + dw * dw;
    #pragma unroll
    for (int o = 16; o >= 1; o >>= 1) s += __shfl_xor(s, o, 32);
    if ((threadIdx.x & 31) == 0) bred[threadIdx.x >> 5] = s;
    __syncthreads();
    if (threadIdx.x == 0) {
        float z = 0.0f;
        #pragma unroll
        for (int i = 0; i < 8; ++i) z += bred[i];
        partial[blockIdx.x] = z;          // deterministic (no float atomics)
    }
}

// Kernel 4: perplexity + loss scalars
__global__ void vq_final(const unsigned int* __restrict__ hist,
                         const float* __restrict__ partial,
                         float* __restrict__ out_loss,
                         float* __restrict__ out_ppl) {
    __shared__ float redA[32], redB[32];
    const int k = threadIdx.x;            // 1024 threads
    float p = (float)hist[k] * (1.0f / 65536.0f);
    float t = p * logf(p + 1e-10f);
    float s = partial[k] + partial[k + 1024] + partial[k + 2048] + partial[k + 3072];
    #pragma unroll
    for (int o = 16; o >= 1; o >>= 1) {
        t += __shfl_xor(t, o, 32);
        s += __shfl_xor(s, o, 32);
    }
    const int lane = k & 31, w = k >> 5;
    if (lane == 0) { redA[w] = t; redB[w] = s; }
    __syncthreads();
    if (k < 32) {
        float tt = redA[k], ss = redB[k];
        #pragma unroll
        for (int o = 16; o >= 1; o >>= 1) {
            tt += __shfl_xor(tt, o, 32);
            ss += __shfl_xor(ss, o, 32);
        }
        if (k == 0) {
            *out_ppl  = expf(-tt);
            *out_loss = 1.25f * ss * (1.0f / 4194304.0f);  // (1 + 0.25) * mean
        }
    }
}

extern "C" void kernel_launch(void* const* d_in, const int* in_sizes, int n_in,
                              void* d_out, int out_size, void* d_ws, size_t ws_size,
                              hipStream_t stream) {
    const float* x   = (const float*)d_in[0];   // (64,64,32,32) f32
    const float* emb = (const float*)d_in[1];   // (1024,64)    f32

    float* outq     = (float*)d_out;            // 4,194,304 quantized + loss + ppl
    float* out_loss = outq + 4194304;
    float* out_ppl  = outq + 4194305;

    char* ws = (char*)d_ws;
    unsigned int* hist    = (unsigned int*)(ws);
    float*        partial = (float*)(ws + 4096);
    float*        enorm   = (float*)(ws + 20480);
    int*          idx     = (int*)(ws + 24576);

    vq_prep  <<<4,    256, 0, stream>>>(emb, enorm, hist);
    vq_argmin<<<512,  256, 0, stream>>>(x, emb, enorm, idx, hist);
    vq_quant <<<4096, 256, 0, stream>>>(x, emb, idx, outq, partial);
    vq_final <<<1,   1024, 0, stream>>>(hist, partial, out_loss, out_ppl);
}